// LinearAttention_38646115729877
// MI455X (gfx1250) — compile-verified
//
#include <hip/hip_runtime.h>
#include <hip/hip_bf16.h>
#include <stdint.h>

#define EPS 1e-6f

typedef __attribute__((ext_vector_type(2))) float v2f;
typedef __attribute__((ext_vector_type(8))) float v8f;

__device__ __forceinline__ float phi(float x) {
    // x > 0 ? x + 1 : exp(x)
    float e = __builtin_amdgcn_exp2f(x * 1.44269504088896340736f);
    return x > 0.0f ? x + 1.0f : e;
}

// ---------------------------------------------------------------------------
// Kernel 1: per-chunk partial sums of phi(k)*m and phi(k)*m * v*m
// ws layout: ws[(bh*C + c)*128 + d]      = sum_k
//            ws[(bh*C + c)*128 + 64 + d] = sum_kv
// ---------------------------------------------------------------------------
__global__ void la_partials(const float* __restrict__ k,
                            const float* __restrict__ v,
                            const float* __restrict__ mask,
                            float* __restrict__ ws,
                            int L, int S, int C, int H) {
    int c  = blockIdx.x;
    int bh = blockIdx.y;
    int b  = bh / H;
    int t  = threadIdx.x;      // 256 threads
    int d  = t & 63;
    int g  = t >> 6;           // 0..3

    const float* kp = k + ((size_t)bh * L + (size_t)c * S) * 64;
    const float* vp = v + ((size_t)bh * L + (size_t)c * S) * 64;
    const float* mp = mask + (size_t)b * L + (size_t)c * S;

    float sk = 0.0f, skv = 0.0f;
    for (int r = g; r < S; r += 4) {
        float m  = mp[r];
        float kf = phi(kp[(size_t)r * 64 + d]) * m;
        float vf = vp[(size_t)r * 64 + d] * m;
        sk  += kf;
        skv += kf * vf;
    }

    __shared__ float red[2][4][64];
    red[0][g][d] = sk;
    red[1][g][d] = skv;
    __syncthreads();
    if (g == 0) {
        float a = red[0][0][d] + red[0][1][d] + red[0][2][d] + red[0][3][d];
        ws[((size_t)bh * C + c) * 128 + d] = a;
    } else if (g == 1) {
        float a = red[1][0][d] + red[1][1][d] + red[1][2][d] + red[1][3][d];
        ws[((size_t)bh * C + c) * 128 + 64 + d] = a;
    }
}

// ---------------------------------------------------------------------------
// Async copy of one 16x64 f32 block (4 KB, contiguous in global) into LDS.
// 8 x global_load_async_to_lds_b128: 32 lanes x 16B = 512B per instruction.
// ---------------------------------------------------------------------------
__device__ __forceinline__ void async_blk(unsigned ldsbase,
                                          const float* kp, const float* vp,
                                          int blk, int buf, unsigned lane) {
    uint64_t ka = (uint64_t)(uintptr_t)(kp + (size_t)blk * 16 * 64);
    uint64_t va = (uint64_t)(uintptr_t)(vp + (size_t)blk * 16 * 64);
    unsigned kl = ldsbase + (unsigned)buf * 4096u;
    unsigned vl = ldsbase + 8192u + (unsigned)buf * 4096u;
#pragma unroll
    for (int i = 0; i < 8; ++i) {
        unsigned off = (unsigned)i * 512u + lane * 16u;
        unsigned la  = kl + off;
        asm volatile("global_load_async_to_lds_b128 %0, %1, %2"
                     :: "v"(la), "v"(off), "s"(ka) : "memory");
    }
#pragma unroll
    for (int i = 0; i < 8; ++i) {
        unsigned off = (unsigned)i * 512u + lane * 16u;
        unsigned la  = vl + off;
        asm volatile("global_load_async_to_lds_b128 %0, %1, %2"
                     :: "v"(la), "v"(off), "s"(va) : "memory");
    }
}

// ---------------------------------------------------------------------------
// Kernel 2: one wave per (b,h,chunk). Scans S rows in 16-row blocks.
// Intra-block inclusive cumsum via V_WMMA_F32_16X16X4_F32 with a constant
// lower-triangular A; carry pre-loaded into the accumulator (replicated over
// rows by the C/D layout). Double-buffered async k/v staging in LDS.
// Two passes per block (K-cumsum/z, then KV-cumsum/out) to keep at most one
// 8-VGPR accumulator live -> no scratch spills.
// ---------------------------------------------------------------------------
__global__ void __launch_bounds__(32)
la_scan(const float* __restrict__ q,
        const float* __restrict__ k,
        const float* __restrict__ v,
        const float* __restrict__ mask,
        const float* __restrict__ ws,
        float* __restrict__ out,
        int L, int S, int C, int H) {
    __shared__ float stage[4096];     // 16 KB: k(buf0,buf1), v(buf0,buf1)

    int c  = blockIdx.x;
    int bh = blockIdx.y;
    int b  = bh / H;
    unsigned lane = threadIdx.x;      // 0..31 (wave32)
    int      half = (int)(lane >> 4); // 0/1
    unsigned ln   = lane & 15u;

    size_t rowbase = (size_t)bh * L + (size_t)c * S;
    const float* kp = k + rowbase * 64;
    const float* vp = v + rowbase * 64;
    const float* qp = q + rowbase * 64;
    float*       op = out + rowbase * 64;
    const float* mp = mask + (size_t)b * L + (size_t)c * S;

    // ---- fold exclusive chunk prefix into carry (lane holds col grp*16+ln) ----
    float ck[4]  = {0.f, 0.f, 0.f, 0.f};
    float ckv[4] = {0.f, 0.f, 0.f, 0.f};
    const float* wsbh = ws + (size_t)bh * C * 128;
    for (int cp = 0; cp < c; ++cp) {
#pragma unroll
        for (int g4 = 0; g4 < 4; ++g4) {
            ck[g4]  += wsbh[cp * 128 + g4 * 16 + ln];
            ckv[g4] += wsbh[cp * 128 + 64 + g4 * 16 + ln];
        }
    }

    // ---- constant triangular A slices: A[m][kcol] = (kcol <= m) ----
    // A 16x4 layout: lane -> M = ln ; VGPR g -> K = 4*s + g + 2*half
    v2f tri[4];
#pragma unroll
    for (int s = 0; s < 4; ++s) {
#pragma unroll
        for (int g = 0; g < 2; ++g) {
            unsigned kcol = (unsigned)(4 * s + g + 2 * half);
            tri[s][g] = (kcol <= ln) ? 1.0f : 0.0f;
        }
    }

    unsigned ldsbase = (unsigned)(uintptr_t)&stage[0];
    int NB = S / 16;
    int bperm_idx = (int)((ln | 16u) << 2);

    async_blk(ldsbase, kp, vp, 0, 0, lane);   // prefetch block 0

    for (int blk = 0; blk < NB; ++blk) {
        int buf = blk & 1;
        if (blk + 1 < NB) {
            async_blk(ldsbase, kp, vp, blk + 1, buf ^ 1, lane);
            asm volatile("s_wait_asynccnt 16" ::: "memory");  // current block done
        } else {
            asm volatile("s_wait_asynccnt 0" ::: "memory");
        }
        const float* kb = &stage[buf * 1024];
        const float* vb = &stage[2048 + buf * 1024];
        int row0 = blk * 16;

        // ---------------- Phase A: K-cumsum tiles -> z partials ----------------
        v8f zp = {0.f, 0.f, 0.f, 0.f, 0.f, 0.f, 0.f, 0.f};
#pragma unroll
        for (int grp = 0; grp < 4; ++grp) {
            v8f accK;
#pragma unroll
            for (int r8 = 0; r8 < 8; ++r8) accK[r8] = ck[grp];
#pragma unroll
            for (int s = 0; s < 4; ++s) {
                // B 4x16 layout: lane -> N = ln ; VGPR g -> K = g + 2*half
                v2f kfB;
#pragma unroll
                for (int g = 0; g < 2; ++g) {
                    int   rl = 4 * s + g + 2 * half;
                    float m  = mp[row0 + rl];
                    kfB[g] = phi(kb[rl * 64 + grp * 16 + ln]) * m;
                }
                accK = __builtin_amdgcn_wmma_f32_16x16x4_f32(false, tri[s], false, kfB,
                                                             (short)0, accK, false, false);
            }
            // zp += phi(q) * sk   (C/D layout: VGPR r8 -> row r8+8*half, lane -> col)
#pragma unroll
            for (int r8 = 0; r8 < 8; ++r8) {
                int row = row0 + r8 + 8 * half;
                zp[r8] += phi(qp[(size_t)row * 64 + grp * 16 + ln]) * accK[r8];
            }
            // carry <- row 15 of inclusive cumsum = VGPR7 on lanes 16..31
            ck[grp] = __int_as_float(__builtin_amdgcn_ds_bpermute(bperm_idx,
                                                                  __float_as_int(accK[7])));
        }

        // z: reduce over the 16 lanes of each half (halves hold different rows)
        v8f inv;
#pragma unroll
        for (int r8 = 0; r8 < 8; ++r8) {
            float zz = zp[r8];
            zz += __int_as_float(__builtin_amdgcn_ds_swizzle(__float_as_int(zz), (1 << 10) | 31));
            zz += __int_as_float(__builtin_amdgcn_ds_swizzle(__float_as_int(zz), (2 << 10) | 31));
            zz += __int_as_float(__builtin_amdgcn_ds_swizzle(__float_as_int(zz), (4 << 10) | 31));
            zz += __int_as_float(__builtin_amdgcn_ds_swizzle(__float_as_int(zz), (8 << 10) | 31));
            float m = mp[row0 + r8 + 8 * half];
            inv[r8] = __builtin_amdgcn_rcpf((zz + EPS) * m);
        }

        // ---------------- Phase B: KV-cumsum tiles -> output ----------------
#pragma unroll
        for (int grp = 0; grp < 4; ++grp) {
            v8f accKV;
#pragma unroll
            for (int r8 = 0; r8 < 8; ++r8) accKV[r8] = ckv[grp];
#pragma unroll
            for (int s = 0; s < 4; ++s) {
                v2f kvB;
#pragma unroll
                for (int g = 0; g < 2; ++g) {
                    int   rl = 4 * s + g + 2 * half;
                    float m  = mp[row0 + rl];
                    float kf = phi(kb[rl * 64 + grp * 16 + ln]) * m;
                    kvB[g]   = kf * (vb[rl * 64 + grp * 16 + ln] * m);
                }
                accKV = __builtin_amdgcn_wmma_f32_16x16x4_f32(false, tri[s], false, kvB,
                                                              (short)0, accKV, false, false);
            }
            // out = phi(q) * kv_cumsum / z   (q reload hits L0/L2)
#pragma unroll
            for (int r8 = 0; r8 < 8; ++r8) {
                int row = row0 + r8 + 8 * half;
                float qf = phi(qp[(size_t)row * 64 + grp * 16 + ln]);
                op[(size_t)row * 64 + grp * 16 + ln] = qf * accKV[r8] * inv[r8];
            }
            ckv[grp] = __int_as_float(__builtin_amdgcn_ds_bpermute(bperm_idx,
                                                                   __float_as_int(accKV[7])));
        }
    }
}

// ---------------------------------------------------------------------------
extern "C" void kernel_launch(void* const* d_in, const int* in_sizes, int n_in,
                              void* d_out, int out_size, void* d_ws, size_t ws_size,
                              hipStream_t stream) {
    const float* q    = (const float*)d_in[0];
    const float* k    = (const float*)d_in[1];
    const float* v    = (const float*)d_in[2];
    const float* mask = (const float*)d_in[3];
    float* out = (float*)d_out;
    float* ws  = (float*)d_ws;

    const int B = 4, H = 16, L = 8192;
    int C = 32;                                   // chunks per (b,h); S = L/C
    while (C > 1 && (size_t)B * H * C * 128 * sizeof(float) > ws_size) C >>= 1;
    int S = L / C;

    dim3 grid(C, B * H, 1);
    if (C > 1) {
        la_partials<<<grid, 256, 0, stream>>>(k, v, mask, ws, L, S, C, H);
    }
    la_scan<<<grid, 32, 0, stream>>>(q, k, v, mask, ws, out, L, S, C, H);
}